// SS2D_76209899701003
// MI455X (gfx1250) — compile-verified
//
#include <hip/hip_runtime.h>
#include <math.h>

// SS2D selective scan for MI455X (gfx1250), fp32 end-to-end.
// Pipeline: [proj: cross-scan gather + WMMA f32 GEMM + dt-proj + softplus]
//        -> [scan: sequential recurrence, exp2-based, 4 lanes per channel]
//        -> [merge + LayerNorm]
// Workspace: delta/ys (aliased) 100,663,296 B + Bs 8,388,608 B + Cs 8,388,608 B = 112 MB.

typedef __attribute__((ext_vector_type(2))) float v2f;
typedef __attribute__((ext_vector_type(8))) float v8f;

#define B_     8
#define D_     192
#define L_     4096
#define K_     4
#define N_     16
#define R_     6
#define C_     38      // R + 2N
#define CPAD   48      // C padded to multiple of 16 for WMMA M-tiles
#define LTILE  128     // l-columns per projection block
#define DCHUNK 48      // K(d)-dim chunk staged in LDS
#define SXS    132     // padded LDS stride for xs tile (bank-conflict mitigation)

// scan-index -> row-major memory position for direction k
__device__ __forceinline__ int pos_map(int k, int l) {
  switch (k) {
    case 0:  return l;                                   // row-major
    case 1:  return ((l & 63) << 6) | (l >> 6);          // transposed (w*H+h -> h*W+w)
    case 2:  return (L_ - 1) - l;                        // reversed row-major
    default: { int j = (L_ - 1) - l;                     // reversed transposed
               return ((j & 63) << 6) | (j >> 6); }
  }
}

// ---------------------------------------------------------------------------
// Kernel 1: cross-scan gather + x_dbl GEMM (WMMA f32 16x16x4) + dt GEMM +
//           softplus. Writes delta (B,K,L,D), Bs (B,K,L,N), Cs (B,K,L,N).
// Grid: B*K*(L/LTILE) = 1024 blocks of 256 threads (8 waves).
// ---------------------------------------------------------------------------
__global__ __launch_bounds__(256) void ss2d_proj(
    const float* __restrict__ x,      // (B,D,H,W)
    const float* __restrict__ Wp,     // (K,C,D)
    const float* __restrict__ dtW,    // (K,D,R)
    const float* __restrict__ dtB,    // (K,D)
    float* __restrict__ delta,        // (B,K,L,D)
    float* __restrict__ Bsg,          // (B,K,L,N)
    float* __restrict__ Csg)          // (B,K,L,N)
{
  __shared__ float sWt[DCHUNK][CPAD];   // W^T chunk: [d][c], c>=38 zero-padded
  __shared__ float sX[DCHUNK][SXS];     // xs chunk:  [d][l]
  __shared__ float sR[CPAD][LTILE];     // x_dbl tile result: [c][l]

  const int tid   = threadIdx.x;
  const int lane  = tid & 31;
  const int wv    = tid >> 5;                 // wave 0..7 -> l-subtile
  const int nblk  = blockIdx.x;
  const int ltile = nblk & 31;                // L_/LTILE = 32
  const int bk    = nblk >> 5;                // 0..31
  const int b     = bk >> 2, k = bk & 3;
  const int l0    = ltile * LTILE;

  const int mrow  = lane & 15;
  const int khalf = (lane >> 4) << 1;         // 0 (lanes 0-15) or 2 (lanes 16-31)

  // 3 accumulators per wave: (m_tile = 0..2, lsub = wv)
  v8f acc[3];
  #pragma unroll
  for (int t = 0; t < 3; ++t)
    #pragma unroll
    for (int j = 0; j < 8; ++j) acc[t][j] = 0.0f;

  for (int ch = 0; ch < D_ / DCHUNK; ++ch) {
    const int d0 = ch * DCHUNK;
    __syncthreads();
    // stage W^T chunk (zero-pad rows c>=38)
    for (int e = tid; e < DCHUNK * CPAD; e += 256) {
      int dd = e % DCHUNK, c = e / DCHUNK;
      sWt[dd][c] = (c < C_) ? Wp[((size_t)(k * C_ + c)) * D_ + d0 + dd] : 0.0f;
    }
    // cross-scan gather of xs chunk
    for (int e = tid; e < DCHUNK * LTILE; e += 256) {
      int l = e % LTILE, dd = e / LTILE;
      sX[dd][l] = x[((size_t)(b * D_ + d0 + dd)) * L_ + pos_map(k, l0 + l)];
    }
    __syncthreads();
    // K-loop: 12 wmma steps of depth 4 over this chunk
    for (int kk = 0; kk < DCHUNK / 4; ++kk) {
      const int dd0 = kk * 4;
      v2f bfrag;  // B[kk][n]: n striped across lanes, K split across lane halves
      bfrag.x = sX[dd0 + khalf    ][wv * 16 + mrow];
      bfrag.y = sX[dd0 + khalf + 1][wv * 16 + mrow];
      #pragma unroll
      for (int t = 0; t < 3; ++t) {
        v2f afrag; // A[m][kk] = W[c=m][d=kk] (ISA 32-bit 16x4 A layout)
        afrag.x = sWt[dd0 + khalf    ][t * 16 + mrow];
        afrag.y = sWt[dd0 + khalf + 1][t * 16 + mrow];
        acc[t] = __builtin_amdgcn_wmma_f32_16x16x4_f32(
            false, afrag, false, bfrag, (short)0, acc[t], false, false);
      }
    }
  }

  __syncthreads();  // all WMMA LDS reads done; sWt/sX reusable
  // spill accumulators to sR; stage dtW + bias into reused LDS
  #pragma unroll
  for (int t = 0; t < 3; ++t)
    #pragma unroll
    for (int j = 0; j < 8; ++j)
      sR[t * 16 + j + ((lane >> 4) << 3)][wv * 16 + mrow] = acc[t][j];
  float* dtw_s  = &sWt[0][0];   // 1152 floats (cap 2304)
  float* bias_s = &sX[0][0];    // 192 floats
  for (int e = tid; e < D_ * R_; e += 256) dtw_s[e] = dtW[(size_t)k * D_ * R_ + e];
  for (int e = tid; e < D_;      e += 256) bias_s[e] = dtB[k * D_ + e];
  __syncthreads();

  // export Bs / Cs (rows R..R+N-1 and R+N..R+2N-1), layout (B,K,L,N)
  for (int e = tid; e < N_ * LTILE; e += 256) {
    int n = e & 15, l = e >> 4;
    size_t off = ((size_t)bk * L_ + l0 + l) * N_ + n;
    Bsg[off] = sR[R_ + n][l];
    Csg[off] = sR[R_ + N_ + n][l];
  }
  // dt projection (K=6, trivial FMA) + softplus -> delta, layout (B,K,L,D)
  for (int e = tid; e < D_ * LTILE; e += 256) {
    int d = e % D_, l = e / D_;
    float a = bias_s[d];
    #pragma unroll
    for (int r = 0; r < R_; ++r)
      a = fmaf(dtw_s[d * R_ + r], sR[r][l], a);
    float sp = (a > 20.0f) ? a : log1pf(__expf(a));
    delta[((size_t)bk * L_ + l0 + l) * D_ + d] = sp;
  }
}

// ---------------------------------------------------------------------------
// Kernel 2: selective scan. One quad of lanes per (b,k,d): 4 states each.
// Reads delta (B,K,L,D), overwrites it in place with ys (safe: same wave,
// read-before-write at each step, element exclusively owned by this quad).
// Grid: B*K*3 = 96 blocks of 256 threads (64 channels per block).
// ---------------------------------------------------------------------------
__global__ __launch_bounds__(256) void ss2d_scan(
    const float* __restrict__ x,       // (B,D,L)
    const float* __restrict__ A_logs,  // (K*D,N)
    const float* __restrict__ Dsg,     // (K*D,)
    const float* __restrict__ Bsg,     // (B,K,L,N)
    const float* __restrict__ Csg,     // (B,K,L,N)
    float* dy)                         // in: delta (B,K,L,D); out: ys
{
  const int tid = threadIdx.x;
  const int ng  = tid & 3;           // state group: n = ng*4 .. ng*4+3
  const int dl  = tid >> 2;          // 0..63
  const int blk = blockIdx.x;
  const int bk  = blk / 3;
  const int d   = (blk % 3) * 64 + dl;
  const int b   = bk >> 2, k = bk & 3;

  // A2 = -exp(A_logs) * log2(e)  so  exp(delta*A) == exp2(delta*A2)
  float A2[4];
  #pragma unroll
  for (int j = 0; j < 4; ++j)
    A2[j] = -__expf(A_logs[(size_t)(k * D_ + d) * N_ + ng * 4 + j]) * 1.4426950408889634f;
  const float dsv = Dsg[k * D_ + d];

  const float*  xrow = x + ((size_t)(b * D_ + d)) * L_;
  const size_t  base = (size_t)bk * L_;
  const float4* Bs4  = (const float4*)Bsg;
  const float4* Cs4  = (const float4*)Csg;

  float h[4] = {0.f, 0.f, 0.f, 0.f};
  for (int s = 0; s < L_; ++s) {
    const float  dv = dy[(base + s) * D_ + d];
    const float  uv = xrow[pos_map(k, s)];
    const float4 bv = Bs4[(base + s) * 4 + ng];
    const float4 cv = Cs4[(base + s) * 4 + ng];
    const float  du = dv * uv;
    const float  bb[4] = {bv.x, bv.y, bv.z, bv.w};
    const float  cc[4] = {cv.x, cv.y, cv.z, cv.w};
    float accv = 0.0f;
    #pragma unroll
    for (int j = 0; j < 4; ++j) {
      float dA = __builtin_amdgcn_exp2f(dv * A2[j]);     // v_exp_f32
      h[j] = fmaf(dA, h[j], du * bb[j]);
      accv = fmaf(h[j], cc[j], accv);
    }
    accv += __shfl_xor(accv, 1, 32);                     // reduce over 4 state
    accv += __shfl_xor(accv, 2, 32);                     // groups in the quad
    if (ng == 0)
      dy[(base + s) * D_ + d] = fmaf(dsv, uv, accv);     // y + D*u, in place
  }
}

// ---------------------------------------------------------------------------
// Kernel 3: cross-merge (undo flips/transpose, sum 4 dirs) + LayerNorm over D.
// One wave per (b,l); contiguous 192-float row reads. Grid: B*L/8 = 4096.
// ---------------------------------------------------------------------------
__global__ __launch_bounds__(256) void ss2d_merge_norm(
    const float* __restrict__ ys,     // (B,K,L,D) (scan-index space)
    const float* __restrict__ gamma,
    const float* __restrict__ beta,
    float* __restrict__ out)          // (B,H,W,D) == (B,L,D)
{
  const int tid  = threadIdx.x;
  const int lane = tid & 31;
  const int wv   = tid >> 5;
  const int gi   = blockIdx.x * 8 + wv;       // 0 .. B*L-1
  const int b    = gi >> 12;
  const int l    = gi & 4095;
  const int lt   = ((l & 63) << 6) | (l >> 6);

  const float* r0 = ys + ((size_t)(b * K_ + 0) * L_ + l)            * D_;
  const float* r1 = ys + ((size_t)(b * K_ + 1) * L_ + lt)           * D_;
  const float* r2 = ys + ((size_t)(b * K_ + 2) * L_ + (L_ - 1 - l)) * D_;
  const float* r3 = ys + ((size_t)(b * K_ + 3) * L_ + (L_ - 1 - lt))* D_;

  float v[6], sum = 0.0f, ssq = 0.0f;
  #pragma unroll
  for (int i = 0; i < 6; ++i) {
    int d = lane + 32 * i;
    float t = (r0[d] + r2[d]) + (r1[d] + r3[d]);
    v[i] = t; sum += t; ssq = fmaf(t, t, ssq);
  }
  #pragma unroll
  for (int off = 16; off; off >>= 1) {
    sum += __shfl_xor(sum, off, 32);
    ssq += __shfl_xor(ssq, off, 32);
  }
  const float mean = sum * (1.0f / 192.0f);
  const float var  = ssq * (1.0f / 192.0f) - mean * mean;
  const float rstd = rsqrtf(var + 1e-5f);
  float* o = out + (size_t)gi * D_;
  #pragma unroll
  for (int i = 0; i < 6; ++i) {
    int d = lane + 32 * i;
    o[d] = fmaf((v[i] - mean) * rstd, gamma[d], beta[d]);
  }
}

// ---------------------------------------------------------------------------
extern "C" void kernel_launch(void* const* d_in, const int* in_sizes, int n_in,
                              void* d_out, int out_size, void* d_ws, size_t ws_size,
                              hipStream_t stream) {
  const float* x      = (const float*)d_in[0];
  const float* Wp     = (const float*)d_in[1];
  const float* dtW    = (const float*)d_in[2];
  const float* dtB    = (const float*)d_in[3];
  const float* A_logs = (const float*)d_in[4];
  const float* Dsg    = (const float*)d_in[5];
  const float* gamma  = (const float*)d_in[6];
  const float* beta   = (const float*)d_in[7];
  float* out = (float*)d_out;

  char* ws = (char*)d_ws;
  float* delta = (float*)ws;                                 // 100,663,296 B (also ys)
  float* Bsg   = (float*)(ws + (size_t)100663296);           //   8,388,608 B
  float* Csg   = (float*)(ws + (size_t)100663296 + 8388608); //   8,388,608 B

  ss2d_proj<<<dim3(B_ * K_ * (L_ / LTILE)), dim3(256), 0, stream>>>(
      x, Wp, dtW, dtB, delta, Bsg, Csg);
  ss2d_scan<<<dim3(B_ * K_ * 3), dim3(256), 0, stream>>>(
      x, A_logs, Dsg, Bsg, Csg, delta);
  ss2d_merge_norm<<<dim3(B_ * L_ / 8), dim3(256), 0, stream>>>(
      delta, gamma, beta, out);
}